// ConvS2SDecoder_89429809037555
// MI455X (gfx1250) — compile-verified
//
#include <hip/hip_runtime.h>
#include <hip/hip_bf16.h>
#include <math.h>

// Problem constants (match reference)
#define V_  32000
#define E_  512
#define H_  512
#define L_  4
#define K_  3
#define B_  16
#define T_  128
#define S_  128
#define M_  (B_ * T_)      // 2048 rows for all GEMMs

typedef float v2f __attribute__((ext_vector_type(2)));
typedef float v8f __attribute__((ext_vector_type(8)));

// ---------------------------------------------------------------------------
// Tiled FP32 WMMA GEMM:  C[M,N] = A[M,K] @ B[K,N] (+ bias[N])
// Block tile 128x64, K-chunk 32, 256 threads = 8 waves.
// Each wave computes a 32x32 tile via 2x2 V_WMMA_F32_16X16X4_F32 accumulators
// (each A/B fragment reused twice -> 1 wmma per 1 aligned ds_load_b64).
// B tile staged TRANSPOSED in LDS so b-fragments are contiguous 8-byte loads.
// All M,N,K divisible by the tile sizes at every call site -> no bounds
// checks -> EXEC is all-ones around every WMMA (ISA requirement).
// ---------------------------------------------------------------------------
#define BM 128
#define BN 64
#define BKC 32
#define PAD 2   // keeps rows 8-byte aligned (34 floats = 136 B)

__global__ __launch_bounds__(256)
void gemm_wmma_f32(const float* __restrict__ A, const float* __restrict__ Bm,
                   const float* __restrict__ bias, float* __restrict__ C,
                   int M, int N, int Kd)
{
    __shared__ float As[BM][BKC + PAD];   // A tile, row-major [m][k]
    __shared__ float Bt[BN][BKC + PAD];   // B tile, TRANSPOSED [n][k]

    const int tid  = threadIdx.x;
    const int wave = tid >> 5;
    const int lane = tid & 31;

    const int bm = blockIdx.y * BM;
    const int bn = blockIdx.x * BN;

    const int wm = (wave >> 1) << 5;   // 0,32,64,96
    const int wn = (wave & 1) << 5;    // 0,32

    const int lrow = lane & 15;        // m (A) / n (B) / n (C,D)
    const int koff = (lane >> 4) << 1; // 0 for lanes 0-15, 2 for lanes 16-31

    v8f acc00 = {0.f,0.f,0.f,0.f,0.f,0.f,0.f,0.f};
    v8f acc01 = acc00, acc10 = acc00, acc11 = acc00;

    for (int kb = 0; kb < Kd; kb += BKC) {
        // Stage A tile 128x32 (coalesced: 32 consecutive k per row)
        #pragma unroll
        for (int i = 0; i < 16; ++i) {
            int idx = tid + i * 256;
            int r = idx >> 5, c = idx & 31;
            As[r][c] = A[(size_t)(bm + r) * Kd + kb + c];
        }
        // Stage B tile 32x64 transposed (coalesced global reads; LDS writes
        // hit 32 distinct banks per wave since 17 is coprime to 32)
        #pragma unroll
        for (int i = 0; i < 8; ++i) {
            int idx = tid + i * 256;
            int r = idx >> 6, c = idx & 63;
            Bt[c][r] = Bm[(size_t)(kb + r) * N + bn + c];
        }
        if (kb + BKC < Kd) {  // uniform condition: no EXEC divergence
            __builtin_prefetch(&A[(size_t)(bm + (tid >> 1)) * Kd + kb + BKC]);
            __builtin_prefetch(&Bm[(size_t)(kb + BKC + (tid >> 3)) * N + bn]);
        }
        __syncthreads();

        #pragma unroll
        for (int kk = 0; kk < BKC; kk += 4) {
            // A frag 16x4 f32: lane l -> m = l%16; VGPR0 = K=2*(l/16), VGPR1 = +1
            // B frag 4x16 f32: VGPRv holds K=v (lanes 0-15) / K=v+2 (lanes 16-31)
            // kk+koff is even and rows are 136 B -> 8-byte aligned ds_load_b64
            const v2f a0 = *(const v2f*)&As[wm + lrow][kk + koff];
            const v2f a1 = *(const v2f*)&As[wm + 16 + lrow][kk + koff];
            const v2f b0 = *(const v2f*)&Bt[wn + lrow][kk + koff];
            const v2f b1 = *(const v2f*)&Bt[wn + 16 + lrow][kk + koff];
            acc00 = __builtin_amdgcn_wmma_f32_16x16x4_f32(false, a0, false, b0,
                                                          (short)0, acc00, false, false);
            acc01 = __builtin_amdgcn_wmma_f32_16x16x4_f32(false, a0, false, b1,
                                                          (short)0, acc01, false, false);
            acc10 = __builtin_amdgcn_wmma_f32_16x16x4_f32(false, a1, false, b0,
                                                          (short)0, acc10, false, false);
            acc11 = __builtin_amdgcn_wmma_f32_16x16x4_f32(false, a1, false, b1,
                                                          (short)0, acc11, false, false);
        }
        __syncthreads();
    }

    // C/D layout: VGPR v, lanes 0-15 -> M=v, lanes 16-31 -> M=v+8; N = lane%16
    const int rb0 = bm + wm + ((lane >> 4) << 3);
    const int rb1 = rb0 + 16;
    const int c0  = bn + wn + lrow;
    const int c1  = c0 + 16;
    const float bb0 = bias ? bias[c0] : 0.0f;
    const float bb1 = bias ? bias[c1] : 0.0f;
    #pragma unroll
    for (int v = 0; v < 8; ++v) {
        C[(size_t)(rb0 + v) * N + c0] = acc00[v] + bb0;
        C[(size_t)(rb0 + v) * N + c1] = acc01[v] + bb1;
        C[(size_t)(rb1 + v) * N + c0] = acc10[v] + bb0;
        C[(size_t)(rb1 + v) * N + c1] = acc11[v] + bb1;
    }
}

// ---------------------------------------------------------------------------
// x[b,t,:] = embed_tokens[tok[b,t],:] + embed_positions[t,:]
// ---------------------------------------------------------------------------
__global__ __launch_bounds__(256)
void embed_kernel(const int* __restrict__ tok, const float* __restrict__ emb,
                  const float* __restrict__ pos, float* __restrict__ x)
{
    const int bt = blockIdx.x;
    const int t  = bt & (T_ - 1);
    const int v  = tok[bt];
    #pragma unroll
    for (int j = 0; j < 2; ++j) {
        int h = threadIdx.x + j * 256;
        x[(size_t)bt * H_ + h] = emb[(size_t)v * E_ + h] + pos[(size_t)t * E_ + h];
    }
}

// ---------------------------------------------------------------------------
// Fused Bahdanau attention for one (b,t):
//   score[s] = sum_h w[h]*tanh(dh[b,t,h]+eh[b,s,h]); softmax over s;
//   x[b,t,:] += attn @ enc[b,:,:]
// Never materializes the [B,T,S,H] tensor (saves ~4.3 GB of HBM traffic;
// eh/enc are 4 MB each and stay resident in the 192 MB L2).
// ---------------------------------------------------------------------------
__global__ __launch_bounds__(128)
void attn_kernel(const float* __restrict__ dh, const float* __restrict__ eh,
                 const float* __restrict__ enc, const unsigned char* __restrict__ mask,
                 const float* __restrict__ w, float* __restrict__ x)
{
    const int bt = blockIdx.x;
    const int b  = bt >> 7;           // T = 128
    const int s  = threadIdx.x;       // one thread per source position

    __shared__ float dh_s[H_];
    __shared__ float w_s[H_];
    __shared__ float red[S_];
    __shared__ float attn_s[S_];

    #pragma unroll
    for (int j = 0; j < 4; ++j) {
        dh_s[s + j * 128] = dh[(size_t)bt * H_ + s + j * 128];
        w_s[s + j * 128]  = w[s + j * 128];
    }
    __syncthreads();

    const float* ehrow = eh + (size_t)(b * S_ + s) * H_;
    float sc = 0.f;
    for (int h = 0; h < H_; ++h)
        sc += w_s[h] * tanhf(dh_s[h] + ehrow[h]);
    if (mask[b * S_ + s]) sc = -INFINITY;

    // softmax over 128 source positions
    red[s] = sc; __syncthreads();
    for (int off = 64; off > 0; off >>= 1) {
        if (s < off) red[s] = fmaxf(red[s], red[s + off]);
        __syncthreads();
    }
    const float mx = red[0]; __syncthreads();
    const float ev = expf(sc - mx);
    red[s] = ev; __syncthreads();
    for (int off = 64; off > 0; off >>= 1) {
        if (s < off) red[s] += red[s + off];
        __syncthreads();
    }
    attn_s[s] = ev / red[0];
    __syncthreads();

    // ctx and residual: each thread handles 4 channels (coalesced over enc rows)
    #pragma unroll
    for (int j = 0; j < 4; ++j) {
        const int e = s + j * 128;
        float cv = 0.f;
        for (int ss = 0; ss < S_; ++ss)
            cv += attn_s[ss] * enc[(size_t)(b * S_ + ss) * E_ + e];
        x[(size_t)bt * H_ + e] += cv;
    }
}

// ---------------------------------------------------------------------------
// Weight-norm repack: Wg[k*H + i, c] = scale[c] * conv_w[c,i,k] / ||conv_w[c,:,:]||
// ---------------------------------------------------------------------------
__global__ __launch_bounds__(256)
void conv_prep_kernel(const float* __restrict__ cw, const float* __restrict__ scale,
                      float* __restrict__ Wg)
{
    const int c   = blockIdx.x;        // out channel 0..2H-1
    const int tid = threadIdx.x;
    __shared__ float red[256];
    const float* row = cw + (size_t)c * (H_ * K_);   // 1536 elems, index = i*K + k
    float ss = 0.f;
    #pragma unroll
    for (int j = 0; j < 6; ++j) { float t = row[tid + j * 256]; ss += t * t; }
    red[tid] = ss; __syncthreads();
    for (int off = 128; off > 0; off >>= 1) {
        if (tid < off) red[tid] += red[tid + off];
        __syncthreads();
    }
    const float rs = scale[c] / sqrtf(red[0]);
    #pragma unroll
    for (int j = 0; j < 6; ++j) {
        int idx = tid + j * 256;          // = i*3 + k
        int i = idx / 3, k = idx - 3 * i;
        Wg[(size_t)(k * H_ + i) * (2 * H_) + c] = row[idx] * rs;
    }
}

// ---------------------------------------------------------------------------
// Causal im2col: xcol[bt, k*H+i] = x[b, t+k-2, i] (zero-padded on the left)
// ---------------------------------------------------------------------------
__global__ __launch_bounds__(256)
void im2col_kernel(const float* __restrict__ x, float* __restrict__ xcol)
{
    const int bt = blockIdx.x;
    const int t  = bt & (T_ - 1);
    #pragma unroll
    for (int j = 0; j < 6; ++j) {
        int idx = threadIdx.x + j * 256;  // 0..1535 = k*512 + i
        int k = idx >> 9, i = idx & 511;
        int tt = t + k - 2;
        float v = (tt >= 0) ? x[(size_t)(bt + k - 2) * H_ + i] : 0.f;
        xcol[(size_t)bt * (K_ * H_) + idx] = v;
    }
}

// ---------------------------------------------------------------------------
// GLU + residual:  x = (y[:, :H] * sigmoid(y[:, H:]) + x) * sqrt(L)
// ---------------------------------------------------------------------------
__global__ __launch_bounds__(256)
void glu_kernel(const float* __restrict__ y, float* __restrict__ x)
{
    const int m = blockIdx.x;
    #pragma unroll
    for (int j = 0; j < 2; ++j) {
        int h = threadIdx.x + j * 256;
        float a = y[(size_t)m * (2 * H_) + h];
        float g = y[(size_t)m * (2 * H_) + H_ + h];
        float sg = 1.0f / (1.0f + expf(-g));
        size_t xi = (size_t)m * H_ + h;
        x[xi] = (a * sg + x[xi]) * 2.0f;   // sqrt(L=4) = 2
    }
}

// ---------------------------------------------------------------------------
extern "C" void kernel_launch(void* const* d_in, const int* in_sizes, int n_in,
                              void* d_out, int out_size, void* d_ws, size_t ws_size,
                              hipStream_t stream) {
    const int*           tok  = (const int*)d_in[0];
    const float*         enc  = (const float*)d_in[1];            // [B,S,E] -> [2048,512]
    const unsigned char* mask = (const unsigned char*)d_in[2];    // bool [B,S]
    const float*         emb  = (const float*)d_in[3];
    const float*         pos  = (const float*)d_in[4];
    const float*         adw  = (const float*)d_in[5];            // [L,H,H]
    const float*         aew  = (const float*)d_in[6];            // [L,E,H]
    const float*         aow  = (const float*)d_in[7];            // [L,H]
    const float*         cw   = (const float*)d_in[8];            // [L,2H,H,K]
    const float*         cs   = (const float*)d_in[9];            // [L,2H]
    const float*         cb   = (const float*)d_in[10];           // [L,2H]
    const float*         pw   = (const float*)d_in[11];           // [H,V]
    const float*         pb   = (const float*)d_in[12];           // [V]
    float* out = (float*)d_out;

    // Workspace carve-up (~38 MB total)
    float* x    = (float*)d_ws;                    // M*H      = 1,048,576
    float* dh   = x    + (size_t)M_ * H_;          // 1,048,576
    float* eh   = dh   + (size_t)M_ * H_;          // 1,048,576
    float* y    = eh   + (size_t)M_ * H_;          // M*2H     = 2,097,152
    float* xcol = y    + (size_t)M_ * 2 * H_;      // M*3H     = 3,145,728
    float* Wg   = xcol + (size_t)M_ * K_ * H_;     // 3H*2H    = 1,572,864

    embed_kernel<<<M_, 256, 0, stream>>>(tok, emb, pos, x);

    for (int l = 0; l < L_; ++l) {
        // dh = x @ attn_dec_w[l]        (2048 x 512 x 512)
        gemm_wmma_f32<<<dim3(H_ / BN, M_ / BM), 256, 0, stream>>>(
            x, adw + (size_t)l * H_ * H_, nullptr, dh, M_, H_, H_);
        // eh = enc @ attn_enc_w[l]      (2048 x 512 x 512)
        gemm_wmma_f32<<<dim3(H_ / BN, M_ / BM), 256, 0, stream>>>(
            enc, aew + (size_t)l * E_ * H_, nullptr, eh, M_, H_, E_);
        // fused tanh-score / softmax / context / residual
        attn_kernel<<<M_, 128, 0, stream>>>(dh, eh, enc, mask, aow + (size_t)l * H_, x);
        // weight-norm repack + im2col + conv-as-GEMM (2048 x 1024 x 1536) + GLU
        conv_prep_kernel<<<2 * H_, 256, 0, stream>>>(
            cw + (size_t)l * 2 * H_ * H_ * K_, cs + (size_t)l * 2 * H_, Wg);
        im2col_kernel<<<M_, 256, 0, stream>>>(x, xcol);
        gemm_wmma_f32<<<dim3(2 * H_ / BN, M_ / BM), 256, 0, stream>>>(
            xcol, Wg, cb + (size_t)l * 2 * H_, y, M_, 2 * H_, K_ * H_);
        glu_kernel<<<M_, 256, 0, stream>>>(y, x);
    }

    // logits = x @ proj_w + proj_b   (2048 x 32000 x 512)
    gemm_wmma_f32<<<dim3(V_ / BN, M_ / BM), 256, 0, stream>>>(x, pw, pb, out, M_, V_, H_);
}